// GraphSim_76175539962371
// MI455X (gfx1250) — compile-verified
//
#include <hip/hip_runtime.h>
#include <hip/hip_bf16.h>

typedef float v2f __attribute__((ext_vector_type(2)));
typedef float v8f __attribute__((ext_vector_type(8)));
typedef unsigned int u32x4 __attribute__((ext_vector_type(4)));
typedef int i32x4 __attribute__((ext_vector_type(4)));
typedef int i32x8 __attribute__((ext_vector_type(8)));

#if defined(__has_builtin)
#if __has_builtin(__builtin_amdgcn_tensor_load_to_lds)
#define HAVE_TDM 1
#endif
#endif

__device__ __forceinline__ v8f wmma_f32x4(v2f a, v2f b, v8f c) {
  // D = A(16x4 f32) * B(4x16 f32) + C(16x16 f32)
  return __builtin_amdgcn_wmma_f32_16x16x4_f32(
      /*neg_a=*/false, a, /*neg_b=*/false, b,
      /*c_mod=*/(short)0, c, /*reuse_a=*/false, /*reuse_b=*/false);
}

#ifdef HAVE_TDM
// Minimal 1-D TDM descriptor: copy `nelems` contiguous f32 from global to LDS.
// D# field packing per CDNA5 ISA ch.8 (group0: count/lds/global/type,
// group1: data_size=4B, tensor_dim0=tile_dim0=nelems, dims1+ = 1).
__device__ __forceinline__ void tdm_load_f32_1d(const void* gsrc, void* lds_dst,
                                                unsigned nelems) {
  unsigned long long ga = (unsigned long long)gsrc;
  unsigned lds = (unsigned)(unsigned long long)lds_dst;   // LDS byte offset
  u32x4 g0;
  g0.x = 1u;                                              // count=1, user D#
  g0.y = lds;                                             // lds_addr [63:32]
  g0.z = (unsigned)(ga & 0xFFFFFFFFull);                  // global_addr[31:0]
  g0.w = (unsigned)((ga >> 32) & 0x01FFFFFFull) | (2u << 30); // [56:32]+type=2
  i32x8 g1;
  g1[0] = (int)(2u << 16);                                // data_size = 4 bytes
  g1[1] = (int)((nelems & 0xFFFFu) << 16);                // tensor_dim0 lo16
  g1[2] = (int)(((nelems >> 16) & 0xFFFFu) | (1u << 16)); // dim0 hi16, dim1=1
  g1[3] = (int)((nelems & 0xFFFFu) << 16);                // tile_dim0
  g1[4] = 1;                                              // tile_dim1=1
  g1[5] = (int)nelems;                                    // dim0_stride lo32
  g1[6] = (int)((nelems & 0xFFFFu) << 16);                // dim1_stride lo16
  g1[7] = (int)((nelems >> 16) & 0xFFFFu);                // dim1_stride hi
  i32x4 z4 = {};
#if __has_include(<hip/amd_detail/amd_gfx1250_TDM.h>)
  i32x8 z8 = {};
  __builtin_amdgcn_tensor_load_to_lds(g0, g1, z4, z4, z8, 0);   // clang-23 form
#else
  __builtin_amdgcn_tensor_load_to_lds(g0, g1, z4, z4, 0);       // ROCm 7.2 form
#endif
}
#endif

// ---------------------------------------------------------------------------
// Kernel 1: fused GCN (3 layers, q & c passes) + per-level similarity GEMM.
// One block per graph. All activations live in LDS (dynamic, ~160KB/WGP).
// ---------------------------------------------------------------------------
__global__ __launch_bounds__(256) void gcn_sims_kernel(
    const float* __restrict__ x_q, const float* __restrict__ x_c,
    const float* __restrict__ w0, const float* __restrict__ b0,
    const float* __restrict__ w1, const float* __restrict__ b1,
    const float* __restrict__ w2, const float* __restrict__ b2,
    const int* __restrict__ eq, const int* __restrict__ ec,
    float* __restrict__ sims)
{
  extern __shared__ float smem[];
  float* Hq  = smem;             // 64*128
  float* Hc  = Hq + 8192;        // 64*128
  float* Fq  = Hc + 8192;        // 64*128 (pre-relu feats)
  float* Fc  = Fq + 8192;
  float* Mbf = Fc + 8192;        // 64*128 GEMM result before scatter
  float* disQ = Mbf + 8192;      // 64
  float* disC = disQ + 64;       // 64

  const int b    = blockIdx.x;
  const int tid  = threadIdx.x;
  const int lane = tid & 31;
  const int wave = tid >> 5;
  const int EB   = 512;          // edges per graph
  const int BE   = 1024 * 512;   // total edges (src row length)
  const int eoff = b * EB;

  // Load node features for this graph.
  for (int i = tid; i < 8192; i += 256) {
    Hq[i] = x_q[(size_t)b * 8192 + i];
    Hc[i] = x_c[(size_t)b * 8192 + i];
  }
  // Degree (in-degree + self loop) -> rsqrt.
  if (tid < 64) { disQ[tid] = 1.0f; disC[tid] = 1.0f; }
  __syncthreads();
  for (int e = tid; e < EB; e += 256) {
    atomicAdd(&disQ[eq[BE + eoff + e] - b * 64], 1.0f);
    atomicAdd(&disC[ec[BE + eoff + e] - b * 64], 1.0f);
  }
  __syncthreads();
  if (tid < 64) { disQ[tid] = rsqrtf(disQ[tid]); disC[tid] = rsqrtf(disC[tid]); }
  __syncthreads();

  const int m0 = lane & 15;          // row/col within 16-wide fragment
  const int kh = (lane >> 4) << 1;   // K sub-offset {0,2}

  for (int l = 0; l < 3; ++l) {
    const float* W    = (l == 0) ? w0 : ((l == 1) ? w1 : w2);
    const float* bias = (l == 0) ? b0 : ((l == 1) ? b1 : b2);

    for (int p = 0; p < 2; ++p) {
      float*       H   = p ? Hc : Hq;
      float*       F   = p ? Fc : Fq;
      const float* dis = p ? disC : disQ;
      const int*   ei  = p ? ec : eq;

      // --- M = H(64x128) @ W(128x128) via f32 WMMA (32 tiles, 4 per wave) ---
      for (int t = wave; t < 32; t += 8) {
        const int tm = (t >> 3) << 4;   // output row tile
        const int tn = (t & 7) << 4;    // output col tile
        v8f acc = {};
        for (int k = 0; k < 128; k += 4) {
          v2f a, bb;
          a.x  = H[(tm + m0) * 128 + k + kh];
          a.y  = H[(tm + m0) * 128 + k + kh + 1];
          bb.x = W[(size_t)(k + kh) * 128 + tn + m0];
          bb.y = W[(size_t)(k + kh + 1) * 128 + tn + m0];
          acc = wmma_f32x4(a, bb, acc);
        }
        const int row0 = tm + (lane >> 4) * 8;
        const int col  = tn + m0;
#pragma unroll
        for (int r = 0; r < 8; ++r) Mbf[(row0 + r) * 128 + col] = acc[r];
      }
      __syncthreads();

      // --- F = bias + self-loop + normalized edge scatter (LDS atomics) ---
      for (int i = tid; i < 8192; i += 256) {
        const float ds = dis[i >> 7];
        F[i] = bias[i & 127] + Mbf[i] * ds * ds;
      }
      __syncthreads();
      for (int e = wave; e < EB; e += 8) {
        const int   s   = ei[eoff + e] - b * 64;
        const int   d   = ei[BE + eoff + e] - b * 64;
        const float nrm = dis[s] * dis[d];
#pragma unroll
        for (int j = 0; j < 4; ++j) {
          const int f = lane * 4 + j;
          atomicAdd(&F[d * 128 + f], Mbf[s * 128 + f] * nrm);
        }
      }
      __syncthreads();
    }

    // --- sims_l[b] = Fq(64x128) @ Fc^T -> 64x64 (16 tiles, 2 per wave) ---
    {
      float* out = sims + ((size_t)l * 1024 + b) * 4096;
      for (int t = wave * 2; t < wave * 2 + 2; ++t) {
        const int tm = (t >> 2) << 4;
        const int tn = (t & 3) << 4;
        v8f acc = {};
        for (int k = 0; k < 128; k += 4) {
          v2f a, bb;
          a.x  = Fq[(tm + m0) * 128 + k + kh];
          a.y  = Fq[(tm + m0) * 128 + k + kh + 1];
          bb.x = Fc[(tn + m0) * 128 + k + kh];      // B[k][n] = Fc[n][k]
          bb.y = Fc[(tn + m0) * 128 + k + kh + 1];
          acc = wmma_f32x4(a, bb, acc);
        }
        const int row0 = tm + (lane >> 4) * 8;
        const int col  = tn + m0;
#pragma unroll
        for (int r = 0; r < 8; ++r) out[(row0 + r) * 64 + col] = acc[r];
      }
    }
    __syncthreads();

    if (l < 2) {
      for (int i = tid; i < 8192; i += 256) {
        Hq[i] = fmaxf(Fq[i], 0.0f);
        Hc[i] = fmaxf(Fc[i], 0.0f);
      }
      __syncthreads();
    }
  }
}

// ---------------------------------------------------------------------------
// Kernel 2: fused conv1(5x5,1->8)+relu+pool2 + conv2(5x5,8->16)+relu+pool2.
// One block per (level, graph). The sims tile is DMA'd into LDS by the
// Tensor Data Mover when available. Writes feat rows in the reference
// concat layout: feat[b, l*16+c, y, x].
// ---------------------------------------------------------------------------
__global__ __launch_bounds__(256) void conv_kernel(
    const float* __restrict__ sims,
    const float* __restrict__ c1w, const float* __restrict__ c1b,
    const float* __restrict__ c2w, const float* __restrict__ c2b,
    float* __restrict__ feat)
{
  __shared__ float S[64 * 64];       // sims tile
  __shared__ float P1[8 * 32 * 32];  // pooled conv1 output
  __shared__ float W1[8 * 25];
  __shared__ float W2[16 * 8 * 25];
  __shared__ float B1[8];
  __shared__ float B2[16];

  const int blk = blockIdx.x;
  const int l   = blk >> 10;
  const int b   = blk & 1023;
  const int tid = threadIdx.x;

  const float* ssrc = sims + ((size_t)l * 1024 + b) * 4096;
#ifdef HAVE_TDM
  if (tid < 32) {                          // one TDM issue per block (wave 0)
    tdm_load_f32_1d(ssrc, S, 4096);
    __builtin_amdgcn_s_wait_tensorcnt((short)0);
  }
#else
  for (int i = tid; i < 4096; i += 256) S[i] = ssrc[i];
#endif
  for (int i = tid; i < 200;  i += 256) W1[i] = c1w[l * 200 + i];
  for (int i = tid; i < 3200; i += 256) W2[i] = c2w[l * 3200 + i];
  if (tid < 8)  B1[tid] = c1b[l * 8 + tid];
  if (tid < 16) B2[tid] = c2b[l * 16 + tid];
  __syncthreads();

  // conv1 + relu + maxpool2 -> P1 (8x32x32)
  for (int idx = tid; idx < 8192; idx += 256) {
    const int c  = idx >> 10;
    const int py = (idx >> 5) & 31;
    const int px = idx & 31;
    const float* wz = &W1[c * 25];
    float mx = -3.0e38f;
    for (int dy = 0; dy < 2; ++dy)
      for (int dx = 0; dx < 2; ++dx) {
        const int oy = py * 2 + dy, ox = px * 2 + dx;
        float sum = B1[c];
#pragma unroll
        for (int ky = 0; ky < 5; ++ky) {
          const int iy = oy + ky - 2;
          if (iy < 0 || iy >= 64) continue;
#pragma unroll
          for (int kx = 0; kx < 5; ++kx) {
            const int ix = ox + kx - 2;
            if (ix < 0 || ix >= 64) continue;
            sum += S[iy * 64 + ix] * wz[ky * 5 + kx];
          }
        }
        mx = fmaxf(mx, fmaxf(sum, 0.0f));
      }
    P1[idx] = mx;
  }
  __syncthreads();

  // conv2 + relu + maxpool2 -> feat (16x16x16 per (l,b))
  for (int idx = tid; idx < 4096; idx += 256) {
    const int c  = idx >> 8;
    const int py = (idx >> 4) & 15;
    const int px = idx & 15;
    float mx = -3.0e38f;
    for (int dy = 0; dy < 2; ++dy)
      for (int dx = 0; dx < 2; ++dx) {
        const int oy = py * 2 + dy, ox = px * 2 + dx;
        float sum = B2[c];
        for (int ci = 0; ci < 8; ++ci) {
          const float* wz = &W2[(c * 8 + ci) * 25];
          const float* In = &P1[ci * 1024];
#pragma unroll
          for (int ky = 0; ky < 5; ++ky) {
            const int iy = oy + ky - 2;
            if (iy < 0 || iy >= 32) continue;
#pragma unroll
            for (int kx = 0; kx < 5; ++kx) {
              const int ix = ox + kx - 2;
              if (ix < 0 || ix >= 32) continue;
              sum += In[iy * 32 + ix] * wz[ky * 5 + kx];
            }
          }
        }
        mx = fmaxf(mx, fmaxf(sum, 0.0f));
      }
    feat[(size_t)b * 12288 + (l * 16 + c) * 256 + py * 16 + px] = mx;
  }
}

// ---------------------------------------------------------------------------
// Kernel 3: h = relu(feat(1024x12288) @ lin_w(12288x512) + lin_b), f32 WMMA.
// Block = 16 rows x 128 cols (8 waves, one 16x16 tile each). The 16x128
// A-chunk is staged in LDS (coalesced) so every wave's A fragments come from
// broadcast ds reads instead of 16-way-strided global loads.
// Grid: 64 row-blocks x 4 col-blocks = 256 blocks.
// ---------------------------------------------------------------------------
__global__ __launch_bounds__(256) void linear_kernel(
    const float* __restrict__ feat, const float* __restrict__ W,
    const float* __restrict__ bias, float* __restrict__ h)
{
  __shared__ float As[16 * 128];            // 8 KB A staging
  const int wave = threadIdx.x >> 5;
  const int lane = threadIdx.x & 31;
  const int rowBlk = blockIdx.x & 63;       // 1024/16 row tiles
  const int colBlk = blockIdx.x >> 6;       // 512/128 col super-tiles
  const int tm = rowBlk << 4;
  const int tn = (colBlk << 7) + (wave << 4);
  const int m0 = lane & 15;
  const int kh = (lane >> 4) << 1;

  v8f acc = {};
  for (int k0 = 0; k0 < 12288; k0 += 128) {
    // Cooperative, coalesced A-chunk load: 16 rows x 128 k.
    for (int i = threadIdx.x; i < 2048; i += 256) {
      const int r = i >> 7, c = i & 127;
      As[i] = feat[(size_t)(tm + r) * 12288 + k0 + c];
    }
    __syncthreads();
    // Prefetch the next A chunk while this one is consumed.
    if (k0 + 128 < 12288) {
      const int r = threadIdx.x >> 4;            // 16 rows
      const int c = (threadIdx.x & 15) << 3;     // 16 x 32B points per row
      __builtin_prefetch(&feat[(size_t)(tm + r) * 12288 + k0 + 128 + c], 0, 1);
    }
#pragma unroll 8
    for (int k = 0; k < 128; k += 4) {
      v2f a, bb;
      a.x  = As[m0 * 128 + k + kh];
      a.y  = As[m0 * 128 + k + kh + 1];
      bb.x = W[(size_t)(k0 + k + kh) * 512 + tn + m0];
      bb.y = W[(size_t)(k0 + k + kh + 1) * 512 + tn + m0];
      acc = wmma_f32x4(a, bb, acc);
    }
    __syncthreads();
  }
  const int   row0 = tm + (lane >> 4) * 8;
  const int   col  = tn + m0;
  const float bc   = bias[col];
#pragma unroll
  for (int r = 0; r < 8; ++r)
    h[(size_t)(row0 + r) * 512 + col] = fmaxf(acc[r] + bc, 0.0f);
}

// ---------------------------------------------------------------------------
// Kernel 4: out[b] = h[b] . score_w + score_b
// ---------------------------------------------------------------------------
__global__ __launch_bounds__(256) void score_kernel(
    const float* __restrict__ h, const float* __restrict__ sw,
    const float* __restrict__ sb, float* __restrict__ out)
{
  __shared__ float red[256];
  const int b = blockIdx.x, tid = threadIdx.x;
  red[tid] = h[(size_t)b * 512 + tid] * sw[tid] +
             h[(size_t)b * 512 + 256 + tid] * sw[256 + tid];
  __syncthreads();
  for (int off = 128; off > 0; off >>= 1) {
    if (tid < off) red[tid] += red[tid + off];
    __syncthreads();
  }
  if (tid == 0) out[b] = red[0] + sb[0];
}

extern "C" void kernel_launch(void* const* d_in, const int* in_sizes, int n_in,
                              void* d_out, int out_size, void* d_ws, size_t ws_size,
                              hipStream_t stream) {
  (void)in_sizes; (void)n_in; (void)out_size; (void)ws_size;
  const float* x_q     = (const float*)d_in[0];
  const float* x_c     = (const float*)d_in[1];
  const float* gcn_w0  = (const float*)d_in[2];
  const float* gcn_b0  = (const float*)d_in[3];
  const float* gcn_w1  = (const float*)d_in[4];
  const float* gcn_b1  = (const float*)d_in[5];
  const float* gcn_w2  = (const float*)d_in[6];
  const float* gcn_b2  = (const float*)d_in[7];
  const float* conv1_w = (const float*)d_in[8];
  const float* conv1_b = (const float*)d_in[9];
  const float* conv2_w = (const float*)d_in[10];
  const float* conv2_b = (const float*)d_in[11];
  const float* lin_w   = (const float*)d_in[12];
  const float* lin_b   = (const float*)d_in[13];
  const float* score_w = (const float*)d_in[14];
  const float* score_b = (const float*)d_in[15];
  const int*   eq      = (const int*)d_in[16];
  const int*   ec      = (const int*)d_in[17];

  float* ws   = (float*)d_ws;
  float* sims = ws;                                  // 3*1024*64*64
  float* feat = sims + (size_t)3 * 1024 * 4096;      // 1024*12288
  float* h    = feat + (size_t)1024 * 12288;         // 1024*512

  const size_t gcn_lds = (size_t)(8192 * 5 + 128) * sizeof(float); // ~160.5 KB

  gcn_sims_kernel<<<1024, 256, gcn_lds, stream>>>(
      x_q, x_c, gcn_w0, gcn_b0, gcn_w1, gcn_b1, gcn_w2, gcn_b2, eq, ec, sims);
  conv_kernel<<<3 * 1024, 256, 0, stream>>>(sims, conv1_w, conv1_b,
                                            conv2_w, conv2_b, feat);
  linear_kernel<<<256, 256, 0, stream>>>(feat, lin_w, lin_b, h);
  score_kernel<<<1024, 256, 0, stream>>>(h, score_w, score_b, (float*)d_out);
}